// GlobalIntrinsicLinear_7404523618628
// MI455X (gfx1250) — compile-verified
//
#include <hip/hip_runtime.h>

// ---------------------------------------------------------------------------
// GlobalIntrinsicLinear for MI455X (gfx1250, wave32, WMMA)
//   out[b,s,o] = sum_i x[b,s,i] * (W0[o,i] + scale * v2[o*768+i]) + bias[o]
//   v2 = FWHT( (FWHT(BB*theta_pad))[Pi] * GG ),  scale = 1/sqrt(DD * sum(GG^2))
// ---------------------------------------------------------------------------

typedef float v2f __attribute__((ext_vector_type(2)));
typedef float v4f __attribute__((ext_vector_type(4)));
typedef float v8f __attribute__((ext_vector_type(8)));

#define LLN   (1 << 20)
#define OUTF  1024
#define INF   768
#define MTOT  16384          // 8 * 2048
#define KTILES (INF / 16)    // 48

// ------------------------------------------------------------------ kernel 1
// First FWHT collapses: only first 2048 entries of BB*theta_pad are nonzero.
// Block b in {0,1} computes A_b = FWHT_1024(BB[b*1024+j]*theta[b*1024+j]).
// Also zeroes the GG^2 accumulator (stream order guarantees it happens first).
__global__ void k_theta_fwht(const float* __restrict__ theta,
                             const float* __restrict__ BB,
                             float* __restrict__ A01,
                             float* __restrict__ sAcc) {
    __shared__ float lds[1024];
    const int t = threadIdx.x;
    const int b = blockIdx.x;
    if (b == 0 && t == 0) sAcc[0] = 0.0f;
    for (int i = t; i < 1024; i += 256) {
        const int gi = b * 1024 + i;
        lds[i] = BB[gi] * theta[gi];
    }
    __syncthreads();
    for (int h = 1; h < 1024; h <<= 1) {
        for (int r = t; r < 512; r += 256) {
            const int i = ((r / h) * (h << 1)) + (r % h);
            const float x = lds[i], y = lds[i + h];
            lds[i] = x + y;
            lds[i + h] = x - y;
        }
        __syncthreads();
    }
    for (int i = t; i < 1024; i += 256) A01[b * 1024 + i] = lds[i];
}

// ------------------------------------------------------------------ kernel 2
// sum(GG^2) via block reduce + one float atomic per block.
__global__ void k_gg_sum(const float* __restrict__ GG, float* __restrict__ sAcc) {
    __shared__ float red[256];
    const int t = threadIdx.x;
    const size_t base = (size_t)blockIdx.x * 1024;
    float p = 0.0f;
    for (int i = 0; i < 4; ++i) {
        const float g = GG[base + t + i * 256];
        p += g * g;
    }
    red[t] = p;
    __syncthreads();
    for (int s = 128; s > 0; s >>= 1) {
        if (t < s) red[t] += red[t + s];
        __syncthreads();
    }
    if (t == 0) atomicAdd(sAcc, red[0]);
}

// ------------------------------------------------------------------ kernel 3
// w[j] = GG[j] * v1[Pi[j]],  v1[r*1024+c] = A0[c] + (-1)^(r&1) * A1[c]
__global__ void k_perm(const int* __restrict__ Pi, const float* __restrict__ GG,
                       const float* __restrict__ A01, float* __restrict__ w) {
    const int j = blockIdx.x * 256 + threadIdx.x;
    const int i = Pi[j];
    const int c = i & 1023;
    const float a0 = A01[c];
    const float a1 = A01[1024 + c];
    const float v = ((i >> 10) & 1) ? (a0 - a1) : (a0 + a1);
    w[j] = GG[j] * v;
}

// ------------------------------------------------------------------ kernel 4
// H_lo: FWHT_1024 over each contiguous 1024-chunk (low 10 index bits).
__global__ void k_fwht_lo(float* __restrict__ w) {
    __shared__ float lds[1024];
    const int t = threadIdx.x;
    const size_t base = (size_t)blockIdx.x * 1024;
    for (int i = t; i < 1024; i += 256) lds[i] = w[base + i];
    __syncthreads();
    for (int h = 1; h < 1024; h <<= 1) {
        for (int r = t; r < 512; r += 256) {
            const int i = ((r / h) * (h << 1)) + (r % h);
            const float x = lds[i], y = lds[i + h];
            lds[i] = x + y;
            lds[i + h] = x - y;
        }
        __syncthreads();
    }
    for (int i = t; i < 1024; i += 256) w[base + i] = lds[i];
}

// ------------------------------------------------------------------ kernel 5
// H_hi: FWHT_1024 along the high 10 bits (stride 1024). 8-column LDS tiles.
__global__ void k_fwht_hi(float* __restrict__ w) {
    __shared__ float lds[1024 * 8];   // 32 KB
    const int t = threadIdx.x;
    const int c = t & 7;
    const int ty = t >> 3;            // 0..31
    const int cb = blockIdx.x * 8;
    for (int r0 = 0; r0 < 1024; r0 += 32) {
        const int row = r0 + ty;
        lds[row * 8 + c] = w[(size_t)row * 1024 + cb + c];
    }
    __syncthreads();
    for (int h = 1; h < 1024; h <<= 1) {
        for (int jj = ty; jj < 512; jj += 32) {
            const int i = ((jj / h) * (h << 1)) + (jj % h);
            const float x = lds[i * 8 + c], y = lds[(i + h) * 8 + c];
            lds[i * 8 + c] = x + y;
            lds[(i + h) * 8 + c] = x - y;
        }
        __syncthreads();
    }
    for (int r0 = 0; r0 < 1024; r0 += 32) {
        const int row = r0 + ty;
        w[(size_t)row * 1024 + cb + c] = lds[row * 8 + c];
    }
}

// ------------------------------------------------------------------ kernel 6
// GEMM: out[m, n] = sum_k x[m,k] * (W0[n,k] + scale*w[n*768+k]) + bias[n]
// Block tile 128(M) x 256(N), 8 waves, wave tile 64x64 via v_wmma_f32_16x16x4_f32.
// Register-staged DOUBLE-BUFFERED LDS: one barrier per K-iteration; global
// loads for tile k+1 issue before the 64-WMMA compute block of tile k.
#define LDA 20
#define LDB 20
__global__ __launch_bounds__(256) void k_gemm(
    const float* __restrict__ x, const float* __restrict__ W0,
    const float* __restrict__ bias, const float* __restrict__ wup,
    const float* __restrict__ sAcc, float* __restrict__ out) {
    __shared__ float As[2][128 * LDA];   // 2 x 10 KB
    __shared__ float Bs[2][256 * LDB];   // 2 x 20 KB
    const int t    = threadIdx.x;
    const int lane = t & 31;
    const int wid  = t >> 5;          // 0..7
    const int wm   = wid & 1;         // M half of block tile
    const int wn   = wid >> 1;        // N quarter of block tile
    const int half = lane >> 4;       // 0: K={0,1}, 1: K={2,3}
    const int lm   = lane & 15;       // row within 16x16 fragment
    const int blockN = blockIdx.x * 256;
    const int blockM = blockIdx.y * 128;
    const float scale = 1.0f / sqrtf(786432.0f * sAcc[0]);

    // Per-thread tile-fill coordinates (loop-invariant).
    const int rowA0 = t >> 2;               // A float4 #t      -> row, kq
    const int kqA0  = (t & 3) << 2;
    const int rowA1 = (t + 256) >> 2;       // A float4 #t+256
    const int kqA1  = ((t + 256) & 3) << 2;

    v8f acc[4][4];
    const v8f zero = {0.f, 0.f, 0.f, 0.f, 0.f, 0.f, 0.f, 0.f};
#pragma unroll
    for (int mt = 0; mt < 4; ++mt)
#pragma unroll
        for (int nt = 0; nt < 4; ++nt) acc[mt][nt] = zero;

    // Staging registers for the in-flight tile.
    v4f sa0, sa1, swv[4], suv[4];

    // ---- prologue: fetch + commit tile 0 into buffer 0
    sa0 = *(const v4f*)(x + (size_t)(blockM + rowA0) * INF + kqA0);
    sa1 = *(const v4f*)(x + (size_t)(blockM + rowA1) * INF + kqA1);
#pragma unroll
    for (int i = 0; i < 4; ++i) {
        const int ff  = t + i * 256;
        const int row = ff >> 2;
        const int kq  = (ff & 3) << 2;
        const size_t gi = (size_t)(blockN + row) * INF + kq;
        swv[i] = *(const v4f*)(W0 + gi);
        suv[i] = *(const v4f*)(wup + gi);
    }
    *(v4f*)(&As[0][rowA0 * LDA + kqA0]) = sa0;
    *(v4f*)(&As[0][rowA1 * LDA + kqA1]) = sa1;
#pragma unroll
    for (int i = 0; i < 4; ++i) {
        const int ff  = t + i * 256;
        const int row = ff >> 2;
        const int kq  = (ff & 3) << 2;
        *(v4f*)(&Bs[0][row * LDB + kq]) = swv[i] + scale * suv[i];
    }
    __syncthreads();

    for (int kt = 0; kt < KTILES; ++kt) {
        const int cur = kt & 1;
        const int nxt = cur ^ 1;
        const bool more = (kt + 1) < KTILES;

        // ---- issue global loads for tile kt+1 (latency hidden by WMMAs)
        if (more) {
            const int kOff = (kt + 1) * 16;
            sa0 = *(const v4f*)(x + (size_t)(blockM + rowA0) * INF + kOff + kqA0);
            sa1 = *(const v4f*)(x + (size_t)(blockM + rowA1) * INF + kOff + kqA1);
#pragma unroll
            for (int i = 0; i < 4; ++i) {
                const int ff  = t + i * 256;
                const int row = ff >> 2;
                const int kq  = (ff & 3) << 2;
                const size_t gi = (size_t)(blockN + row) * INF + kOff + kq;
                swv[i] = *(const v4f*)(W0 + gi);
                suv[i] = *(const v4f*)(wup + gi);
            }
        }

        // ---- 64 WMMAs on the current buffer
        const float* __restrict__ Ac = As[cur];
        const float* __restrict__ Bc = Bs[cur];
#pragma unroll
        for (int s = 0; s < 4; ++s) {
            const int kk = (s << 2) + (half << 1);
            v2f af[4], bf[4];
#pragma unroll
            for (int mt = 0; mt < 4; ++mt)
                af[mt] = *(const v2f*)(Ac + (wm * 64 + mt * 16 + lm) * LDA + kk);
#pragma unroll
            for (int nt = 0; nt < 4; ++nt)
                bf[nt] = *(const v2f*)(Bc + (wn * 64 + nt * 16 + lm) * LDB + kk);
#pragma unroll
            for (int mt = 0; mt < 4; ++mt)
#pragma unroll
                for (int nt = 0; nt < 4; ++nt)
                    acc[mt][nt] = __builtin_amdgcn_wmma_f32_16x16x4_f32(
                        false, af[mt], false, bf[nt], (short)0, acc[mt][nt],
                        false, false);
        }

        // ---- commit staged tile kt+1 into the other buffer
        if (more) {
            *(v4f*)(&As[nxt][rowA0 * LDA + kqA0]) = sa0;
            *(v4f*)(&As[nxt][rowA1 * LDA + kqA1]) = sa1;
#pragma unroll
            for (int i = 0; i < 4; ++i) {
                const int ff  = t + i * 256;
                const int row = ff >> 2;
                const int kq  = (ff & 3) << 2;
                *(v4f*)(&Bs[nxt][row * LDB + kq]) = swv[i] + scale * suv[i];
            }
        }
        __syncthreads();
    }

    // --- epilogue: C layout VGPR v -> M = v + 8*half, N = lm (per ISA 7.12.2)
#pragma unroll
    for (int nt = 0; nt < 4; ++nt) {
        const int col = blockN + wn * 64 + nt * 16 + lm;
        const float bv = bias[col];
#pragma unroll
        for (int mt = 0; mt < 4; ++mt) {
            const int rowBase = blockM + wm * 64 + mt * 16 + half * 8;
#pragma unroll
            for (int v = 0; v < 8; ++v)
                out[(size_t)(rowBase + v) * OUTF + col] = acc[mt][nt][v] + bv;
        }
    }
}

// ---------------------------------------------------------------------------
extern "C" void kernel_launch(void* const* d_in, const int* in_sizes, int n_in,
                              void* d_out, int out_size, void* d_ws, size_t ws_size,
                              hipStream_t stream) {
    (void)in_sizes; (void)n_in; (void)out_size; (void)ws_size;
    const float* x     = (const float*)d_in[0];  // (8,2048,768)
    const float* theta = (const float*)d_in[1];  // (2048,)
    const float* W0    = (const float*)d_in[2];  // (1024,768)
    const float* bias  = (const float*)d_in[3];  // (1024,)
    const float* BB    = (const float*)d_in[4];  // (2^20,)
    const float* GG    = (const float*)d_in[5];  // (2^20,)
    const int*   Pi    = (const int*)d_in[6];    // (2^20,)
    float* out = (float*)d_out;                  // (8,2048,1024)

    float* sAcc = (float*)d_ws;                  // 1 float
    float* A01  = (float*)d_ws + 64;             // 2048 floats
    float* w    = (float*)d_ws + 64 + 2048;      // 2^20 floats (16B aligned)

    k_theta_fwht<<<2,    256, 0, stream>>>(theta, BB, A01, sAcc);
    k_gg_sum   <<<1024,  256, 0, stream>>>(GG, sAcc);
    k_perm     <<<4096,  256, 0, stream>>>(Pi, GG, A01, w);
    k_fwht_lo  <<<1024,  256, 0, stream>>>(w);
    k_fwht_hi  <<<128,   256, 0, stream>>>(w);
    k_gemm     <<<dim3(OUTF / 256, MTOT / 128), 256, 0, stream>>>(
        x, W0, bias, w, sAcc, out);
}